// GINEModel_59004260713105
// MI455X (gfx1250) — compile-verified
//
#include <hip/hip_runtime.h>
#include <hip/hip_bf16.h>

// ---------------- problem constants (match reference) ----------------
#define N_NODES 50000
#define N_EDGES 800000
#define IN_DIM  32
#define H_DIM   64
#define ED_DIM  4
#define L_LAYERS 3
#define G_GRAPHS 512
#define OUT_DIM 2
#define BN_EPS  1e-5f
#define PN_EPS  1e-5f

typedef __attribute__((ext_vector_type(2))) float v2f;
typedef __attribute__((ext_vector_type(8))) float v8f;

// fp32 WMMA: D(16x16) = A(16x4) * B(4x16) + C   (exact fp32, wave32)
__device__ __forceinline__ v8f wmma4(v2f a, v2f b, v8f c) {
    return __builtin_amdgcn_wmma_f32_16x16x4_f32(
        /*neg_a=*/false, a, /*neg_b=*/false, b,
        /*c_mod=*/(short)0, c, /*reuse_a=*/false, /*reuse_b=*/false);
}

// =====================================================================
// h = x @ W_in + b      x:[N,32]  W:[32,64]  -> h:[N,64]
// one wave per 16-row tile; 4 column sub-tiles of 16; K=32 in steps of 4
// =====================================================================
__global__ void k_lin_in(const float* __restrict__ x,
                         const float* __restrict__ W,
                         const float* __restrict__ bias,
                         float* __restrict__ h) {
    const int lane = threadIdx.x & 31;
    const int tile = blockIdx.x * (blockDim.x >> 5) + (threadIdx.x >> 5);
    if (tile >= N_NODES / 16) return;          // wave-uniform exit, EXEC stays all-1s
    const int row0 = tile * 16;
    const int mn  = lane & 15;                 // A row in tile / B col in ntile
    const int khi = lane >> 4;                 // K half-select

    v8f acc[4] = {};
    const float* xrow = x + (row0 + mn) * IN_DIM;
#pragma unroll
    for (int k0 = 0; k0 < IN_DIM; k0 += 4) {
        const int kk = k0 + 2 * khi;
        v2f a = *(const v2f*)(xrow + kk);
#pragma unroll
        for (int nt = 0; nt < 4; ++nt) {
            v2f b;
            b.x = W[ kk      * H_DIM + nt * 16 + mn];
            b.y = W[(kk + 1) * H_DIM + nt * 16 + mn];
            acc[nt] = wmma4(a, b, acc[nt]);
        }
    }
#pragma unroll
    for (int nt = 0; nt < 4; ++nt) {
        const int col = nt * 16 + mn;
        const float bv = bias[col];
#pragma unroll
        for (int r = 0; r < 8; ++r) {
            const int row = row0 + r + 8 * khi;   // D layout: VGPR r -> M=r (+8 for hi half)
            h[row * H_DIM + col] = acc[nt][r] + bv;
        }
    }
}

// =====================================================================
// GEMM (K=64) + linear bias + BatchNorm(eval) + ReLU.
// If A1 != null:   A = (1+eps[li])*A0 + A1   (GINE combine), else A = A0.
// out = relu( ((A@W + lb) - rm) * g * rsqrt(rv+eps) + bb )
// =====================================================================
__global__ void k_gemm64_bn_relu(const float* __restrict__ A0,
                                 const float* __restrict__ A1,
                                 const float* __restrict__ epsp, int li,
                                 const float* __restrict__ W,
                                 const float* __restrict__ lb,
                                 const float* __restrict__ g,
                                 const float* __restrict__ bb,
                                 const float* __restrict__ rm,
                                 const float* __restrict__ rv,
                                 float* __restrict__ outp) {
    const int lane = threadIdx.x & 31;
    const int tile = blockIdx.x * (blockDim.x >> 5) + (threadIdx.x >> 5);
    if (tile >= N_NODES / 16) return;
    const int row0 = tile * 16;
    const int mn  = lane & 15;
    const int khi = lane >> 4;

    const bool comb = (A1 != nullptr);
    const float epl = comb ? (1.0f + epsp[li]) : 1.0f;
    const int rowoff = (row0 + mn) * H_DIM;

    v8f acc[4] = {};
#pragma unroll
    for (int k0 = 0; k0 < H_DIM; k0 += 4) {
        const int kk = k0 + 2 * khi;
        v2f a = *(const v2f*)(A0 + rowoff + kk);
        if (comb) {
            v2f a1 = *(const v2f*)(A1 + rowoff + kk);
            a.x = epl * a.x + a1.x;
            a.y = epl * a.y + a1.y;
        }
#pragma unroll
        for (int nt = 0; nt < 4; ++nt) {
            v2f b;
            b.x = W[ kk      * H_DIM + nt * 16 + mn];
            b.y = W[(kk + 1) * H_DIM + nt * 16 + mn];
            acc[nt] = wmma4(a, b, acc[nt]);
        }
    }
#pragma unroll
    for (int nt = 0; nt < 4; ++nt) {
        const int col = nt * 16 + mn;
        const float scale = g[col] * __frsqrt_rn(rv[col] + BN_EPS);
        const float base  = lb[col] - rm[col];
        const float shift = bb[col];
#pragma unroll
        for (int r = 0; r < 8; ++r) {
            const int row = row0 + r + 8 * khi;
            float v = (acc[nt][r] + base) * scale + shift;
            outp[row * H_DIM + col] = v > 0.0f ? v : 0.0f;
        }
    }
}

// =====================================================================
// Edge phase: m = relu(h[src] + edge_attr@Wd + bd); atomic scatter to agg[dst]
// 8 threads per edge, 8 columns each.  (memory/atomic bound — the roofline)
// =====================================================================
__global__ void k_edge(const int* __restrict__ ei,
                       const float* __restrict__ ea,
                       const float* __restrict__ eW,   // [4,64] this layer
                       const float* __restrict__ eb,   // [64]
                       const float* __restrict__ h,
                       float* __restrict__ agg) {
    const long tid = (long)blockIdx.x * blockDim.x + threadIdx.x;
    if (tid >= (long)N_EDGES * 8) return;
    const int e  = (int)(tid >> 3);
    const int j0 = ((int)tid & 7) * 8;
    const int src = ei[e];
    const int dst = ei[N_EDGES + e];
    const float a0 = ea[e * 4 + 0], a1 = ea[e * 4 + 1];
    const float a2 = ea[e * 4 + 2], a3 = ea[e * 4 + 3];
    const float* hs = h + src * H_DIM + j0;
    float* ag = agg + dst * H_DIM + j0;
#pragma unroll
    for (int j = 0; j < 8; ++j) {
        const int c = j0 + j;
        float w = eb[c] + a0 * eW[c] + a1 * eW[64 + c] + a2 * eW[128 + c] + a3 * eW[192 + c];
        float m = hs[j] + w;
        m = m > 0.0f ? m : 0.0f;
        atomicAdd(ag + j, m);
    }
}

// ---------------- segment helpers ----------------
__global__ void k_counts(const int* __restrict__ batch, float* __restrict__ counts) {
    int n = blockIdx.x * blockDim.x + threadIdx.x;
    if (n >= N_NODES) return;
    atomicAdd(&counts[batch[n]], 1.0f);
}

// dst[batch[n]*64 + j] += src[n*64 + j]   (used for PairNorm mean and final pool)
__global__ void k_seg_scatter(const float* __restrict__ src, const int* __restrict__ batch,
                              float* __restrict__ dst) {
    int tid = blockIdx.x * blockDim.x + threadIdx.x;
    if (tid >= N_NODES * H_DIM) return;
    const int n = tid >> 6, j = tid & 63;
    atomicAdd(&dst[batch[n] * H_DIM + j], src[tid]);
}

// per-node squared deviation, wave32 shuffle-reduce, atomic into ss[g]
__global__ void k_pn_var(const float* __restrict__ h, const int* __restrict__ batch,
                         const float* __restrict__ meansum, const float* __restrict__ counts,
                         float* __restrict__ ss) {
    const int node = blockIdx.x * (blockDim.x >> 5) + (threadIdx.x >> 5);
    if (node >= N_NODES) return;
    const int lane = threadIdx.x & 31;
    const int g = batch[node];
    const float c = fmaxf(counts[g], 1.0f);
    float s = 0.0f;
#pragma unroll
    for (int j = lane; j < H_DIM; j += 32) {
        float hc = h[node * H_DIM + j] - meansum[g * H_DIM + j] / c;
        s += hc * hc;
    }
#pragma unroll
    for (int off = 16; off > 0; off >>= 1) s += __shfl_xor(s, off, 32);
    if (lane == 0) atomicAdd(&ss[g], s);
}

__global__ void k_pn_apply(float* __restrict__ h, const int* __restrict__ batch,
                           const float* __restrict__ meansum, const float* __restrict__ counts,
                           const float* __restrict__ ss) {
    int tid = blockIdx.x * blockDim.x + threadIdx.x;
    if (tid >= N_NODES * H_DIM) return;
    const int n = tid >> 6, j = tid & 63;
    const int g = batch[n];
    const float c = fmaxf(counts[g], 1.0f);
    const float hc = h[tid] - meansum[g * H_DIM + j] / c;
    h[tid] = hc / sqrtf(PN_EPS + ss[g] / c);   // PN_SCALE == 1
}

// out[g,:] = pooled[g,:] @ cls_W + cls_b     (512 x 64 x 2 — trivial)
__global__ void k_cls(const float* __restrict__ pooled, const float* __restrict__ cW,
                      const float* __restrict__ cb, float* __restrict__ out) {
    int g = blockIdx.x * blockDim.x + threadIdx.x;
    if (g >= G_GRAPHS) return;
    float s0 = cb[0], s1 = cb[1];
#pragma unroll 8
    for (int j = 0; j < H_DIM; ++j) {
        const float p = pooled[g * H_DIM + j];
        s0 += p * cW[j * 2 + 0];
        s1 += p * cW[j * 2 + 1];
    }
    out[g * 2 + 0] = s0;
    out[g * 2 + 1] = s1;
}

// =====================================================================
extern "C" void kernel_launch(void* const* d_in, const int* in_sizes, int n_in,
                              void* d_out, int out_size, void* d_ws, size_t ws_size,
                              hipStream_t stream) {
    const float* x        = (const float*)d_in[0];
    const int*   ei       = (const int*)  d_in[1];
    const float* ea       = (const float*)d_in[2];
    const int*   batch    = (const int*)  d_in[3];
    const float* lin_in_W = (const float*)d_in[4];
    const float* lin_in_b = (const float*)d_in[5];
    const float* edge_W   = (const float*)d_in[6];
    const float* edge_b   = (const float*)d_in[7];
    const float* mlp_W1   = (const float*)d_in[8];
    const float* mlp_b1   = (const float*)d_in[9];
    const float* mbn_g    = (const float*)d_in[10];
    const float* mbn_b    = (const float*)d_in[11];
    const float* mbn_rm   = (const float*)d_in[12];
    const float* mbn_rv   = (const float*)d_in[13];
    const float* mlp_W2   = (const float*)d_in[14];
    const float* mlp_b2   = (const float*)d_in[15];
    const float* epsp     = (const float*)d_in[16];
    const float* bn_g     = (const float*)d_in[17];
    const float* bn_b     = (const float*)d_in[18];
    const float* bn_rm    = (const float*)d_in[19];
    const float* bn_rv    = (const float*)d_in[20];
    const float* cls_W    = (const float*)d_in[21];
    const float* cls_b    = (const float*)d_in[22];
    float* out = (float*)d_out;

    // ---- workspace layout (floats) ----
    const size_t NH = (size_t)N_NODES * H_DIM;       // 3.2M
    const size_t GH = (size_t)G_GRAPHS * H_DIM;      // 32768
    float* ws      = (float*)d_ws;
    float* h       = ws;
    float* agg     = h + NH;
    float* tmp     = agg + NH;
    float* meansum = tmp + NH;
    float* ss      = meansum + GH;
    float* counts  = ss + G_GRAPHS;
    float* pooled  = counts + G_GRAPHS;              // total ~38.7 MB

    const int TPB = 256;
    const int gTiles   = (N_NODES / 16 + 7) / 8;           // waves per block = 8
    const int gEdges   = (int)(((long)N_EDGES * 8 + TPB - 1) / TPB);
    const int gNodeCol = (N_NODES * H_DIM + TPB - 1) / TPB;
    const int gNodes   = (N_NODES + TPB - 1) / TPB;
    const int gWaveNode = (N_NODES + 7) / 8;

    // counts per graph (once per call)
    hipMemsetAsync(counts, 0, G_GRAPHS * sizeof(float), stream);
    k_counts<<<gNodes, TPB, 0, stream>>>(batch, counts);

    // h = x @ W_in + b   (WMMA fp32)
    k_lin_in<<<gTiles, TPB, 0, stream>>>(x, lin_in_W, lin_in_b, h);

    for (int i = 0; i < L_LAYERS; ++i) {
        // GINE message + scatter-add
        hipMemsetAsync(agg, 0, NH * sizeof(float), stream);
        k_edge<<<gEdges, TPB, 0, stream>>>(ei, ea,
                                           edge_W + (size_t)i * ED_DIM * H_DIM,
                                           edge_b + (size_t)i * H_DIM, h, agg);

        // tmp = relu(BN1((1+eps)h + agg) @ W1 + b1))   (WMMA fp32)
        k_gemm64_bn_relu<<<gTiles, TPB, 0, stream>>>(
            h, agg, epsp, i,
            mlp_W1 + (size_t)i * H_DIM * H_DIM, mlp_b1 + (size_t)i * H_DIM,
            mbn_g + (size_t)i * H_DIM, mbn_b + (size_t)i * H_DIM,
            mbn_rm + (size_t)i * H_DIM, mbn_rv + (size_t)i * H_DIM, tmp);

        // h = relu(BN(tmp @ W2 + b2))                  (WMMA fp32)
        k_gemm64_bn_relu<<<gTiles, TPB, 0, stream>>>(
            tmp, nullptr, nullptr, 0,
            mlp_W2 + (size_t)i * H_DIM * H_DIM, mlp_b2 + (size_t)i * H_DIM,
            bn_g + (size_t)i * H_DIM, bn_b + (size_t)i * H_DIM,
            bn_rm + (size_t)i * H_DIM, bn_rv + (size_t)i * H_DIM, h);

        // PairNorm per graph segment (layers 0,1)
        if (i < L_LAYERS - 1) {
            hipMemsetAsync(meansum, 0, GH * sizeof(float), stream);
            hipMemsetAsync(ss, 0, G_GRAPHS * sizeof(float), stream);
            k_seg_scatter<<<gNodeCol, TPB, 0, stream>>>(h, batch, meansum);
            k_pn_var<<<gWaveNode, TPB, 0, stream>>>(h, batch, meansum, counts, ss);
            k_pn_apply<<<gNodeCol, TPB, 0, stream>>>(h, batch, meansum, counts, ss);
        }
    }

    // global add pool + classifier
    hipMemsetAsync(pooled, 0, GH * sizeof(float), stream);
    k_seg_scatter<<<gNodeCol, TPB, 0, stream>>>(h, batch, pooled);
    k_cls<<<(G_GRAPHS + TPB - 1) / TPB, TPB, 0, stream>>>(pooled, cls_W, cls_b, out);
}